// CASCADES_v8_ResonantCore_48095043780872
// MI455X (gfx1250) — compile-verified
//
#include <hip/hip_runtime.h>
#include <cmath>

// CDNA5 / gfx1250, wave32. Shapes hardcoded from the reference:
// B=4, S=4096, D=4096, R=8, K=4, all float32.
#define B_  4
#define S_  4096
#define D_  4096
#define R_  8
#define K_  4
#define EPSF 1e-8f
#define TEMPF 0.05f

#define DSPLIT 4
#define DCHUNK (D_ / DSPLIT)  // 1024

typedef __attribute__((ext_vector_type(2))) float v2f;
typedef __attribute__((ext_vector_type(8))) float v8f;

// ---------------------------------------------------------------------------
// Pass 0: zero-padded V:  Vpad[n][d] = (n < 8) ? V[n][d] : 0   (16 x D)
// Removes all EXEC divergence from the pass-1 B-fragment loads.
// ---------------------------------------------------------------------------
__global__ __launch_bounds__(256) void cascades_pass0_vpad(
    const float* __restrict__ V, float* __restrict__ Vpad) {
  const int idx = blockIdx.x * 256 + threadIdx.x;  // 0 .. 16*D-1
  const int n = idx >> 12;                         // D_ == 4096
  Vpad[idx] = (n < R_) ? V[idx] : 0.0f;
}

// ---------------------------------------------------------------------------
// Pass 1: xV[b,s,r] += sum_{d in chunk} x[b,s,d] * V[r,d]  (WMMA f32 16x16x4)
//         colsum[b,d] = sum_s x[b,s,d]                     (ds_add_f32)
// One wave owns 16 flattened rows (b*S+s) and a D/4 chunk of the reduction;
// block = 8 waves = 128 rows. Grid = (B*S/128) * DSPLIT = 512 blocks.
// xV and colsum are accumulated with global_atomic_add_f32 (L2-resident).
// ---------------------------------------------------------------------------
__global__ __launch_bounds__(256) void cascades_pass1_xv_colsum(
    const float* __restrict__ x, const float* __restrict__ Vpad,
    float* __restrict__ xV, float* __restrict__ colsum) {
  // Per-wave 16x32 staging tile, row stride 36 dwords (36*r mod 64 distinct
  // for r<16 -> conflict-free strided A-fragment reads). 8*16*36 floats.
  __shared__ float tile[8 * 16 * 36];
  __shared__ float csum[DCHUNK];  // block column sums for its d-slice (4 KB)

  const int tid  = threadIdx.x;
  const int lane = tid & 31;
  const int w    = tid >> 5;
  const int rb   = blockIdx.x / DSPLIT;      // row-block
  const int dsb  = blockIdx.x % DSPLIT;      // d-chunk
  const int dlo  = dsb * DCHUNK;
  const int blockRow = rb * 128;             // flattened b*S + s base
  const int b        = blockRow >> 12;       // S_ == 4096
  const int waveRow  = blockRow + w * 16;

  for (int i = tid; i < DCHUNK; i += 256) csum[i] = 0.0f;
  __syncthreads();

  float* tw = &tile[w * (16 * 36)];
  // A-fragment lane mapping (ISA 7.12.2, 32-bit A 16x4):
  //   lanes 0-15:  M=lane,    K = {0,1}
  //   lanes 16-31: M=lane-16, K = {2,3}
  const int aRow  = lane & 15;
  const int khalf = (lane >> 4) * 2;   // 0 or 2
  const int nCol  = lane & 15;         // B/C fragment column

  v8f acc = {};  // C fragment: xV tile (only N<8 columns meaningful)

  const int colq  = (lane & 7) * 4;    // 0..28, step 4
  const int rbase = lane >> 3;         // 0..3

  for (int d0 = dlo; d0 < dlo + DCHUNK; d0 += 32) {
    // ---- stage 16x32 of x into LDS (coalesced float4 loads) ----
    float4 sum4 = make_float4(0.f, 0.f, 0.f, 0.f);
#pragma unroll
    for (int j = 0; j < 4; ++j) {
      const int row = rbase + j * 4;   // 0..15
      const size_t gidx = (size_t)(waveRow + row) * D_ + (size_t)d0 + colq;
      const float4 v = *(const float4*)&x[gidx];
      __builtin_prefetch(&x[gidx + 32], 0, 3);  // next d-slab of same row
      *(float4*)&tw[row * 36 + colq] = v;
      sum4.x += v.x; sum4.y += v.y; sum4.z += v.z; sum4.w += v.w;
    }
    // column-sum partials (4 lanes share each column; LDS float atomics)
    const int cb = d0 - dlo + colq;
    unsafeAtomicAdd(&csum[cb + 0], sum4.x);
    unsafeAtomicAdd(&csum[cb + 1], sum4.y);
    unsafeAtomicAdd(&csum[cb + 2], sum4.z);
    unsafeAtomicAdd(&csum[cb + 3], sum4.w);

    // ---- 8 K=4 chunks of WMMA over this 32-wide d slab (branch-free) ----
#pragma unroll
    for (int kk = 0; kk < 8; ++kk) {
      const v2f a  = *(const v2f*)&tw[aRow * 36 + kk * 4 + khalf];
      const v2f bf = *(const v2f*)&Vpad[(size_t)nCol * D_ + d0 + kk * 4 + khalf];
      acc = __builtin_amdgcn_wmma_f32_16x16x4_f32(
          false, a, false, bf, (short)0, acc, false, false);
    }
  }

  // C layout: VGPR v -> M = v (+8 for lanes>=16), N = lane&15.
  // Accumulate the D-chunk partial into xV (zero-initialized by launcher).
  if (nCol < R_) {
    const int mOff = (lane < 16) ? 0 : 8;
#pragma unroll
    for (int v = 0; v < 8; ++v)
      unsafeAtomicAdd(&xV[(size_t)(waveRow + v + mOff) * R_ + nCol], acc[v]);
  }

  __syncthreads();
  for (int i = tid; i < DCHUNK; i += 256)
    unsafeAtomicAdd(&colsum[(size_t)b * D_ + dlo + i], csum[i]);
}

// ---------------------------------------------------------------------------
// Pass 2a: routing. One block per batch. centroid -> cosine sim -> softmax
//          -> Lam[b] = sum_k w_k * core_pool[k]  (8x8)
// ---------------------------------------------------------------------------
__global__ __launch_bounds__(256) void cascades_pass2a_route(
    const float* __restrict__ x, const float* __restrict__ colsum,
    const float* __restrict__ keys, const float* __restrict__ pool,
    float* __restrict__ Lam) {
  const int b = blockIdx.x, tid = threadIdx.x;
  float cc = 0.f, dk[K_] = {0.f, 0.f, 0.f, 0.f}, qq[K_] = {0.f, 0.f, 0.f, 0.f};
  const float inv_s = 0.3f / (float)S_;
  for (int d = tid; d < D_; d += 256) {
    const float c = 0.7f * x[((size_t)b * S_ + (S_ - 1)) * D_ + d] +
                    inv_s * colsum[(size_t)b * D_ + d];
    cc += c * c;
#pragma unroll
    for (int k = 0; k < K_; ++k) {
      const float kv = keys[(size_t)k * D_ + d];
      dk[k] += c * kv;
      qq[k] += kv * kv;
    }
  }
  __shared__ float red[9][8];
  float vals[9] = {cc, dk[0], dk[1], dk[2], dk[3], qq[0], qq[1], qq[2], qq[3]};
#pragma unroll
  for (int i = 0; i < 9; ++i) {
    float v = vals[i];
    for (int off = 16; off > 0; off >>= 1) v += __shfl_down(v, off, 32);
    if ((tid & 31) == 0) red[i][tid >> 5] = v;
  }
  __syncthreads();
  __shared__ float wgt[K_];
  if (tid == 0) {
    float t[9];
#pragma unroll
    for (int i = 0; i < 9; ++i) {
      float s = 0.f;
      for (int j = 0; j < 8; ++j) s += red[i][j];
      t[i] = s;
    }
    const float cn = fmaxf(sqrtf(t[0]), EPSF);
    float sim[K_], m = -1e30f;
#pragma unroll
    for (int k = 0; k < K_; ++k) {
      sim[k] = (t[1 + k] / (cn * fmaxf(sqrtf(t[5 + k]), EPSF))) / TEMPF;
      m = fmaxf(m, sim[k]);
    }
    float e[K_], sum = 0.f;
#pragma unroll
    for (int k = 0; k < K_; ++k) { e[k] = __expf(sim[k] - m); sum += e[k]; }
#pragma unroll
    for (int k = 0; k < K_; ++k) wgt[k] = e[k] / sum;
  }
  __syncthreads();
  if (tid < R_ * R_) {
    float a = 0.f;
#pragma unroll
    for (int k = 0; k < K_; ++k) a += wgt[k] * pool[k * (R_ * R_) + tid];
    Lam[b * (R_ * R_) + tid] = a;  // Lam[b][i][j]
  }
}

// ---------------------------------------------------------------------------
// Pass 2b: fold Lam into U:  W[b,d,r] = sum_j U[d,j] * Lam[b,j,r]
// (so out = xV @ W[b]^T; removes the xVL pass entirely). Grid = B*D/256.
// ---------------------------------------------------------------------------
__global__ __launch_bounds__(256) void cascades_pass2b_foldW(
    const float* __restrict__ U, const float* __restrict__ Lam,
    float* __restrict__ W) {
  const int idx = blockIdx.x * 256 + threadIdx.x;  // b*D + d
  const int b = idx >> 12;
  const float* u = &U[(size_t)(idx & (D_ - 1)) * R_];
  const float* L = &Lam[b * (R_ * R_)];
  float o[R_];
#pragma unroll
  for (int r = 0; r < R_; ++r) {
    float a = 0.f;
#pragma unroll
    for (int j = 0; j < R_; ++j) a += u[j] * L[j * R_ + r];
    o[r] = a;
  }
  *(float4*)&W[(size_t)idx * R_ + 0] = make_float4(o[0], o[1], o[2], o[3]);
  *(float4*)&W[(size_t)idx * R_ + 4] = make_float4(o[4], o[5], o[6], o[7]);
}

// ---------------------------------------------------------------------------
// Pass 3: out[b,s,d] = sum_r xV[b,s,r] * W[b,d,r]
// 16x16 output tile per WMMA pair (two K=4 chunks over r). One wave handles
// one 16-row s-tile x 256-wide d-group (A fragments loaded once).
// Grid = (B*S/16 tiles * 16 d-groups) / 8 waves = 2048 blocks.
// ---------------------------------------------------------------------------
__global__ __launch_bounds__(256) void cascades_pass3_out(
    const float* __restrict__ xV, const float* __restrict__ W,
    float* __restrict__ out) {
  const int tid  = threadIdx.x;
  const int lane = tid & 31;
  const int w    = tid >> 5;
  const int gw   = blockIdx.x * 8 + w;   // 0..16383
  const int sTile  = gw >> 4;            // 0..1023
  const int dGroup = gw & 15;
  const int row0 = sTile * 16;           // flattened b*S + s
  const int b    = row0 >> 12;
  const int nCol  = lane & 15;
  const int khalf = (lane >> 4) * 2;
  const int aRow  = lane & 15;

  // A fragments: xV rows, K = r-chunk {0..3} and {4..7}
  const v2f a0 = *(const v2f*)&xV[(size_t)(row0 + aRow) * R_ + 0 + khalf];
  const v2f a1 = *(const v2f*)&xV[(size_t)(row0 + aRow) * R_ + 4 + khalf];

  const int dBase = dGroup * 256;
  const int mOff  = (lane < 16) ? 0 : 8;
  for (int dt = 0; dt < 16; ++dt) {
    const int d0 = dBase + dt * 16;
    const size_t wb = ((size_t)b * D_ + d0 + nCol) * R_;
    const v2f b0 = *(const v2f*)&W[wb + 0 + khalf];
    const v2f b1 = *(const v2f*)&W[wb + 4 + khalf];
    v8f c = {};
    c = __builtin_amdgcn_wmma_f32_16x16x4_f32(false, a0, false, b0, (short)0,
                                              c, false, false);
    c = __builtin_amdgcn_wmma_f32_16x16x4_f32(false, a1, false, b1, (short)0,
                                              c, false, false);
#pragma unroll
    for (int v = 0; v < 8; ++v)
      out[(size_t)(row0 + v + mOff) * D_ + d0 + nCol] = c[v];
  }
}

// ---------------------------------------------------------------------------
// Host-side launcher (graph-capture safe: only async ops on `stream`).
// ws layout (floats):
//   colsum[B*D] | xV[B*S*R] | Lam[B*R*R] | W[B*D*R] | Vpad[16*D]
// colsum+xV are contiguous and zeroed with one hipMemsetAsync.
// ---------------------------------------------------------------------------
extern "C" void kernel_launch(void* const* d_in, const int* in_sizes, int n_in,
                              void* d_out, int out_size, void* d_ws,
                              size_t ws_size, hipStream_t stream) {
  const float* x    = (const float*)d_in[0];
  const float* V    = (const float*)d_in[1];
  const float* U    = (const float*)d_in[2];
  const float* pool = (const float*)d_in[3];
  const float* keys = (const float*)d_in[4];
  float* out = (float*)d_out;

  float* ws     = (float*)d_ws;
  float* colsum = ws;                                   // 16384
  float* xV     = ws + (B_ * D_);                       // 131072
  float* Lam    = xV + (B_ * S_ * R_);                  // 256
  float* W      = Lam + (B_ * R_ * R_);                 // 131072
  float* Vpad   = W + (B_ * D_ * R_);                   // 65536

  hipMemsetAsync(colsum, 0,
                 (size_t)(B_ * D_ + B_ * S_ * R_) * sizeof(float), stream);

  cascades_pass0_vpad<<<(16 * D_) / 256, 256, 0, stream>>>(V, Vpad);
  cascades_pass1_xv_colsum<<<(B_ * S_ / 128) * DSPLIT, 256, 0, stream>>>(
      x, Vpad, xV, colsum);
  cascades_pass2a_route<<<B_, 256, 0, stream>>>(x, colsum, keys, pool, Lam);
  cascades_pass2b_foldW<<<(B_ * D_) / 256, 256, 0, stream>>>(U, Lam, W);
  cascades_pass3_out<<<2048, 256, 0, stream>>>(xV, W, out);
}